// GCN_Layer_68350109549100
// MI455X (gfx1250) — compile-verified
//
#include <hip/hip_runtime.h>
#include <hip/hip_bf16.h>

// ---------------------------------------------------------------------------
// LightGCN-style pipeline for MI455X (gfx1250, wave32).
//   Phase 1: edge SpMM (wave-per-edge atomic scatter) + row-L2-norm accumulate
//   Phase 2: dense linears via V_WMMA_F32_16X16X4_F32 (fp32-exact path)
//   Phase 3: batch x poi scoring GEMM, 5 N-tiles per wave (A-fragment reuse,
//            5 independent accumulator chains for WMMA pipelining)
// All WMMA grids are exact (dims are multiples of the tile sizes) so every
// WMMA wave runs with EXEC == all ones, as the ISA requires.
// ---------------------------------------------------------------------------

typedef __attribute__((ext_vector_type(2))) float v2f;
typedef __attribute__((ext_vector_type(8))) float v8f;

#define EMB 128

// ---------------- simple utility kernels ----------------
__global__ void copy_f32(float* __restrict__ dst, const float* __restrict__ src, int n) {
    int i = blockIdx.x * blockDim.x + threadIdx.x;
    if (i < n) dst[i] = src[i];
}

__global__ void zero_f32(float* __restrict__ dst, int n) {
    int i = blockIdx.x * blockDim.x + threadIdx.x;
    if (i < n) dst[i] = 0.0f;
}

// ---------------- SpMM: out[dIdx[e]] += val[e] * src[sIdx[e]] ----------------
// One wave per edge: edge metadata is read once per wave, each lane handles
// 4 dims (b128 gather + 4x global_atomic_add_f32 scatter). Operand tables are
// L2-resident (10-26 MB << 192 MB), so cost is dominated by the atomics.
__global__ void spmm_scatter(float* __restrict__ out, const float* __restrict__ src,
                             const int* __restrict__ dIdx, const int* __restrict__ sIdx,
                             const float* __restrict__ val, int nEdges) {
    int lane = threadIdx.x & 31;
    int e = blockIdx.x * (blockDim.x >> 5) + (threadIdx.x >> 5);
    if (e >= nEdges) return;
    const int   row = dIdx[e];
    const int   col = sIdx[e];
    const float v   = val[e];
    const float4 s = *(const float4*)(src + (size_t)col * EMB + lane * 4);
    float* o = out + (size_t)row * EMB + lane * 4;
    atomicAdd(o + 0, v * s.x);
    atomicAdd(o + 1, v * s.y);
    atomicAdd(o + 2, v * s.z);
    atomicAdd(o + 3, v * s.w);
}

// ---------------- sum[row] += feat[row] / ||feat[row]||_2 ----------------
// One wave per row; lane handles 4 dims (float4); wave32 xor-shuffle reduce.
__global__ void norm_accum(float* __restrict__ sum, const float* __restrict__ feat, int nRows) {
    int lane = threadIdx.x & 31;
    int row = blockIdx.x * (blockDim.x >> 5) + (threadIdx.x >> 5);
    if (row >= nRows) return;
    const float4 v = *(const float4*)(feat + (size_t)row * EMB + lane * 4);
    float ss = v.x * v.x + v.y * v.y + v.z * v.z + v.w * v.w;
    #pragma unroll
    for (int off = 16; off >= 1; off >>= 1)
        ss += __shfl_xor(ss, off, 32);
    float inv = 1.0f / sqrtf(ss);
    float4* s = (float4*)(sum + (size_t)row * EMB + lane * 4);
    float4 cur = *s;
    cur.x += v.x * inv; cur.y += v.y * inv; cur.z += v.z * inv; cur.w += v.w * inv;
    *s = cur;
}

// ---------------- out = X @ W^T  (X: nRows x 128, W: 128 x 128) ----------------
// One wave per 16x16 output tile, 8 waves/block, K=128 via 32 x WMMA f32 16x16x4.
// A lane fragment: row M=lane%16, K pair at kb=(lane>=16?2:0)  -> contiguous float2.
// B lane fragment: B[k][n] = W[n][k]  -> same contiguous float2 off W's row n.
__global__ void __launch_bounds__(256) gemm_wmma_xWT(const float* __restrict__ X,
                                                     const float* __restrict__ W,
                                                     float* __restrict__ out) {
    const int lane = threadIdx.x & 31;
    const int warp = threadIdx.x >> 5;
    const int nTilesN = EMB / 16;                      // 8
    int tile = blockIdx.x * 8 + warp;                  // exact grid, no guard
    int m0 = (tile / nTilesN) << 4;
    int n0 = (tile % nTilesN) << 4;
    const int r  = lane & 15;
    const int kb = (lane >> 4) << 1;                   // 0 or 2
    const float* xrow = X + (size_t)(m0 + r) * EMB + kb;
    const float* wrow = W + (size_t)(n0 + r) * EMB + kb;
    v8f acc = {};
    #pragma unroll 8
    for (int k0 = 0; k0 < EMB; k0 += 4) {
        v2f a = *(const v2f*)(xrow + k0);
        v2f b = *(const v2f*)(wrow + k0);
        acc = __builtin_amdgcn_wmma_f32_16x16x4_f32(false, a, false, b,
                                                    (short)0, acc, false, false);
    }
    const int col = n0 + r;
    const int rowBase = m0 + ((lane >> 4) << 3);       // +8 for lanes 16..31
    #pragma unroll
    for (int i = 0; i < 8; ++i)
        out[(size_t)(rowBase + i) * EMB + col] = acc[i];
}

// ---------------- gather batch rows of user_feature ----------------
__global__ void gather_rows(float* __restrict__ dst, const float* __restrict__ src,
                            const int* __restrict__ idx) {
    dst[(size_t)blockIdx.x * EMB + threadIdx.x] =
        src[(size_t)idx[blockIdx.x] * EMB + threadIdx.x];
}

// ---------------- up[b,p] = dot(UB[b], PF[p])  (2048 x 20000, K=128) ----------------
// Each wave computes a 16x80 strip: 5 N-tiles sharing one A fragment per K-step
// (5x less A traffic) with 5 independent accumulators (no serial WMMA C-chain).
// PF (10 MB) is fully L2-resident and reused 128x across M tiles.
#define NT 5
__global__ void __launch_bounds__(256) gemm_wmma_score(const float* __restrict__ UB,
                                                       const float* __restrict__ PF,
                                                       float* __restrict__ out,
                                                       int nPois) {
    const int lane = threadIdx.x & 31;
    const int warp = threadIdx.x >> 5;
    const int nGroupsN = nPois / (16 * NT);            // 250
    int wid = blockIdx.x * 8 + warp;                   // exact grid, no guard
    int m0 = (wid / nGroupsN) << 4;
    int n0 = (wid % nGroupsN) * (16 * NT);
    const int r  = lane & 15;
    const int kb = (lane >> 4) << 1;                   // 0 or 2
    const float* xrow = UB + (size_t)(m0 + r) * EMB + kb;
    const float* prow = PF + (size_t)(n0 + r) * EMB + kb;
    v8f acc[NT] = {};
    #pragma unroll 4
    for (int k0 = 0; k0 < EMB; k0 += 4) {
        v2f a = *(const v2f*)(xrow + k0);
        #pragma unroll
        for (int t = 0; t < NT; ++t) {
            v2f b = *(const v2f*)(prow + (size_t)t * 16 * EMB + k0);
            acc[t] = __builtin_amdgcn_wmma_f32_16x16x4_f32(false, a, false, b,
                                                           (short)0, acc[t], false, false);
        }
    }
    const int rowBase = m0 + ((lane >> 4) << 3);       // +8 for lanes 16..31
    #pragma unroll
    for (int t = 0; t < NT; ++t) {
        const int col = n0 + t * 16 + r;
        #pragma unroll
        for (int i = 0; i < 8; ++i)
            out[(size_t)(rowBase + i) * nPois + col] = acc[t][i];
    }
}

// ---------------------------------------------------------------------------
extern "C" void kernel_launch(void* const* d_in, const int* in_sizes, int n_in,
                              void* d_out, int out_size, void* d_ws, size_t ws_size,
                              hipStream_t stream) {
    const int*   edge_row   = (const int*)d_in[0];
    const int*   edge_col   = (const int*)d_in[1];
    const float* edge_val   = (const float*)d_in[2];
    const float* user_embed = (const float*)d_in[3];
    const float* poi_embed  = (const float*)d_in[4];
    const int*   batch_user = (const int*)d_in[5];
    const float* W_user     = (const float*)d_in[6];
    const float* W_poi      = (const float*)d_in[7];

    const int nEdges = in_sizes[0];
    const int nUsers = in_sizes[3] / EMB;   // 50000
    const int nPois  = in_sizes[4] / EMB;   // 20000
    const int nBatch = in_sizes[5];         // 2048
    const int nLayers = 3;                  // reference constant

    // workspace layout (floats)
    float* ws    = (float*)d_ws;
    float* u_sum = ws;
    float* p_sum = u_sum + (size_t)nUsers * EMB;
    float* lu    = p_sum + (size_t)nPois * EMB;
    float* lp    = lu    + (size_t)nUsers * EMB;
    float* poiF  = lp    + (size_t)nPois * EMB;
    float* UB    = poiF  + (size_t)nPois * EMB;

    float* up    = (float*)d_out;                       // [nBatch, nPois]
    float* userF = up + (size_t)nBatch * nPois;         // [nUsers, EMB]

    const int nu = nUsers * EMB;
    const int np = nPois * EMB;

    copy_f32<<<(nu + 255) / 256, 256, 0, stream>>>(u_sum, user_embed, nu);
    copy_f32<<<(np + 255) / 256, 256, 0, stream>>>(p_sum, poi_embed, np);
    copy_f32<<<(np + 255) / 256, 256, 0, stream>>>(lp,    poi_embed, np);

    const int edgeBlocks = (nEdges + 7) / 8;            // 8 waves/block, 1 edge/wave

    for (int L = 0; L < nLayers; ++L) {
        // user <- A @ poi
        zero_f32<<<(nu + 255) / 256, 256, 0, stream>>>(lu, nu);
        spmm_scatter<<<edgeBlocks, 256, 0, stream>>>(lu, lp, edge_row, edge_col,
                                                     edge_val, nEdges);
        norm_accum<<<(nUsers + 7) / 8, 256, 0, stream>>>(u_sum, lu, nUsers);
        // poi <- A^T @ user (unnormalized lu)
        zero_f32<<<(np + 255) / 256, 256, 0, stream>>>(lp, np);
        spmm_scatter<<<edgeBlocks, 256, 0, stream>>>(lp, lu, edge_col, edge_row,
                                                     edge_val, nEdges);
        norm_accum<<<(nPois + 7) / 8, 256, 0, stream>>>(p_sum, lp, nPois);
    }

    // dense linears: blocks = nRows/16 (8 N-tiles * 8 waves/block exactly)
    gemm_wmma_xWT<<<nUsers / 16, 256, 0, stream>>>(u_sum, W_user, userF);
    gemm_wmma_xWT<<<nPois / 16,  256, 0, stream>>>(p_sum, W_poi, poiF);

    // batch gather + scoring GEMM (16x80 strip per wave)
    gather_rows<<<nBatch, EMB, 0, stream>>>(UB, userF, batch_user);
    const int scoreWaves = (nBatch / 16) * (nPois / (16 * NT));   // 128 * 250 = 32000
    gemm_wmma_score<<<scoreWaves / 8, 256, 0, stream>>>(UB, poiF, up, nPois);
}